// DMax_34076270526484
// MI455X (gfx1250) — compile-verified
//
#include <hip/hip_runtime.h>
#include <hip/hip_bf16.h>
#include <stdint.h>

// Segment-prefix-max: out[s, :] = max over rows [start_s, start_s + size_s - win + 1) of x[:, 128]
// Streaming-bound kernel (512 MB read @ 23.3 TB/s ~= 22 us roofline):
// stage 32-row x 128-col f32 tiles into LDS with the CDNA5 Tensor Data Mover
// (tensor_load_to_lds, up to 3 tiles in flight on TENSORcnt), reduce from LDS
// with ds_load_b128 + v_max_f32, one workgroup barrier per tile.

#define DCOLS      128
#define BDIM       256
#define TILE_ROWS  32
#define TILE_BYTES (TILE_ROWS * DCOLS * 4)     // 16 KB
#define NBUF       4                           // pipeline depth 3
#define SCRATCH_OFF (NBUF * TILE_BYTES)        // 64 KB
#define SMEM_BYTES (SCRATCH_OFF + BDIM * 16)   // 68 KB

typedef __attribute__((ext_vector_type(4))) unsigned int v4u;
typedef __attribute__((ext_vector_type(8))) int          v8i;
typedef __attribute__((ext_vector_type(4))) int          v4i;

__device__ __forceinline__ float4 f4max(float4 a, float4 b) {
    float4 r;
    r.x = fmaxf(a.x, b.x);
    r.y = fmaxf(a.y, b.y);
    r.z = fmaxf(a.z, b.z);
    r.w = fmaxf(a.w, b.w);
    return r;
}

// Issue one TDM load of a [rows x 128] f32 tile (row-major, row stride 128
// elements) from global byte address `gaddr` into LDS byte offset `lds_addr`.
__device__ __forceinline__ void tdm_load_tile(uint32_t lds_addr, uint64_t gaddr,
                                              uint32_t rows) {
    // D# group 0 (128b): count=1 | lds_addr[63:32] | global_addr[120:64] | type=2
    v4u g0;
    g0.x = 1u;                                   // count=1, is_restore=0, gather off
    g0.y = lds_addr;                             // LDS byte address
    g0.z = (uint32_t)gaddr;                      // global_addr[31:0]
    g0.w = ((uint32_t)(gaddr >> 32) & 0x01FFFFFFu) | (2u << 30); // addr[56:32] | type=2

    // D# group 1 (256b)
    v8i g1;
    g1[0] = (int)(2u << 16);                     // workgroup_mask=0, data_size=2 (4B)
    g1[1] = (int)((uint32_t)DCOLS << 16);        // tensor_dim0[15:0]=128 (bits [79:48] low half)
    g1[2] = (int)((rows & 0xFFFFu) << 16);       // tensor_dim1[15:0]=rows
    g1[3] = (int)((uint32_t)DCOLS << 16);        // tensor_dim1[31:16]=0 | tile_dim0=128
    g1[4] = (int)(rows & 0xFFFFu);               // tile_dim1=rows | tile_dim2=0
    g1[5] = DCOLS;                               // tensor_dim0_stride[31:0]=128 elements
    g1[6] = 0;                                   // stride0 hi | tensor_dim1_stride lo
    g1[7] = 0;

    // Groups 2/3: 2D tile; keep higher tensor dims benign (=1), tiles unused (=0)
    v4i g2; g2[0] = 1; g2[1] = 1; g2[2] = 0; g2[3] = 0;
    v4i g3; g3[0] = 0; g3[1] = (int)(1u << 16); g3[2] = 0; g3[3] = 0;

    // clang-23 / amdgpu-toolchain form: 6 args, trailing int32x8 zero-filled.
    v8i g4; g4[0] = 0; g4[1] = 0; g4[2] = 0; g4[3] = 0;
            g4[4] = 0; g4[5] = 0; g4[6] = 0; g4[7] = 0;

    __builtin_amdgcn_tensor_load_to_lds(g0, g1, g2, g3, g4, /*cpol=*/0);
}

__global__ __launch_bounds__(BDIM) void DMax_segmax_kernel(
    const float* __restrict__ x, const long long* __restrict__ sizes,
    const int* __restrict__ winp, float* __restrict__ out, int nseg) {
    __shared__ __align__(16) unsigned char smem[SMEM_BYTES];

    const int t    = threadIdx.x;
    const int seg  = blockIdx.x;
    const int col4 = t & 31;     // which float4 of the 128-col row
    const int rgrp = t >> 5;     // row group 0..7

    // --- exclusive prefix sum of sizes[0..seg) -> segment start row ---
    long long psum = 0;
    for (int i = t; i < seg; i += BDIM) psum += sizes[i];
    long long* red = (long long*)(smem + SCRATCH_OFF);
    red[t] = psum;
    __syncthreads();
    for (int off = BDIM / 2; off > 0; off >>= 1) {
        if (t < off) red[t] += red[t + off];
        __syncthreads();
    }
    const long long start = red[0];
    __syncthreads();   // scratch region reused later

    const long long sz     = sizes[seg];
    const int       win    = winp[0];
    const long long validL = sz - (long long)win + 1;
    float* oseg = out + (size_t)seg * DCOLS;

    if (validL <= 0) {
        if (t < 32) {
            float4 ninf = make_float4(-INFINITY, -INFINITY, -INFINITY, -INFINITY);
            *(float4*)(oseg + col4 * 4) = ninf;
        }
        return;
    }
    const int valid = (int)validL;

    const uint64_t gbase = (uint64_t)(uintptr_t)x +
                           (uint64_t)start * (uint64_t)(DCOLS * 4);
    // LDS byte offset of our shared block (flat LDS pointers carry offset in low bits)
    const uint32_t lds_base = (uint32_t)(uintptr_t)(void*)smem;

    float4 acc = make_float4(-INFINITY, -INFINITY, -INFINITY, -INFINITY);
    const int ntile = (valid + TILE_ROWS - 1) / TILE_ROWS;

    // Prologue: wave 0 issues up to NBUF-1 tiles (TDM ignores EXEC; one wave issues)
    if (rgrp == 0) {
        const int npre = min(NBUF - 1, ntile);
        for (int k = 0; k < npre; ++k) {
            const int rk = min(TILE_ROWS, valid - k * TILE_ROWS);
            tdm_load_tile(lds_base + (uint32_t)(k * TILE_BYTES),
                          gbase + (uint64_t)k * (uint64_t)(TILE_ROWS * DCOLS * 4),
                          (uint32_t)rk);
        }
    }

    for (int ti = 0; ti < ntile; ++ti) {
        const int rows = min(TILE_ROWS, valid - ti * TILE_ROWS);

        // Wait until tile ti is complete: outstanding-after = min(NBUF-2, ntile-ti-1)
        if (rgrp == 0) {
            const int rem = ntile - ti - 1;
            if (rem >= 2)      __builtin_amdgcn_s_wait_tensorcnt(2);
            else if (rem == 1) __builtin_amdgcn_s_wait_tensorcnt(1);
            else               __builtin_amdgcn_s_wait_tensorcnt(0);
        }
        // Publishes tile ti to all waves AND proves tile ti-1 (same buffer as
        // ti+NBUF-1) has been fully consumed -> safe to issue the overwrite.
        __syncthreads();

        if (rgrp == 0 && ti + NBUF - 1 < ntile) {
            const int k  = ti + NBUF - 1;
            const int rk = min(TILE_ROWS, valid - k * TILE_ROWS);
            tdm_load_tile(lds_base + (uint32_t)((k & (NBUF - 1)) * TILE_BYTES),
                          gbase + (uint64_t)k * (uint64_t)(TILE_ROWS * DCOLS * 4),
                          (uint32_t)rk);
        }

        const unsigned char* buf = smem + (ti & (NBUF - 1)) * TILE_BYTES;
        const unsigned char* p   = buf + rgrp * (DCOLS * 4) + col4 * 16;
        if (rows == TILE_ROWS) {
            float4 v0 = *(const float4*)(p);
            float4 v1 = *(const float4*)(p +  8 * (DCOLS * 4));
            float4 v2 = *(const float4*)(p + 16 * (DCOLS * 4));
            float4 v3 = *(const float4*)(p + 24 * (DCOLS * 4));
            acc = f4max(acc, f4max(f4max(v0, v1), f4max(v2, v3)));
        } else {
            for (int r = rgrp; r < rows; r += 8) {
                float4 v = *(const float4*)(buf + r * (DCOLS * 4) + col4 * 16);
                acc = f4max(acc, v);
            }
        }
    }

    // --- combine the 8 row-groups per float4 column ---
    float4* part = (float4*)(smem + SCRATCH_OFF);
    part[t] = acc;
    __syncthreads();
    if (t < 32) {
        float4 m = part[t];
#pragma unroll
        for (int g = 1; g < 8; ++g) m = f4max(m, part[g * 32 + t]);
        *(float4*)(oseg + t * 4) = m;
    }
}

extern "C" void kernel_launch(void* const* d_in, const int* in_sizes, int n_in,
                              void* d_out, int out_size, void* d_ws, size_t ws_size,
                              hipStream_t stream) {
    const float*     x     = (const float*)d_in[0];
    const long long* sizes = (const long long*)d_in[1];   // jnp.int64
    const int*       winp  = (const int*)d_in[2];         // scalar (low 32 bits)
    float*           out   = (float*)d_out;
    const int n = in_sizes[1];                            // number of segments

    DMax_segmax_kernel<<<dim3(n), dim3(BDIM), 0, stream>>>(x, sizes, winp, out, n);
}